// ProjectedGaussianRasterizer_89386859364996
// MI455X (gfx1250) — compile-verified
//
#include <hip/hip_runtime.h>
#include <stdint.h>

// ---------------------------------------------------------------------------
// Gaussian splat rasterizer for MI455X (gfx1250, wave32).
// Compute-bound (AI ~1e4 FLOP/byte): depth-sort + tile culling + early exit,
// sigma evaluated on the matrix pipe as a K=6 bilinear form via
// V_WMMA_F32_16X16X4_F32; double-buffered async global->LDS chunk staging.
// ---------------------------------------------------------------------------

#define G_N 8192
#define W_IMG 512
#define H_IMG 512
#define TILE 16
#define CHUNK 256
#define RECF 16               // floats per sorted gaussian record (64B)
#define SIG_STRIDE 17         // padded LDS stride (conflict-free: gcd(17,64)=1)
#define ALPHA_THRESH (1.0f/255.0f)
#define TRANS_THRESH 1.0e-4f
#define ALPHA_CAP 0.99f

typedef __attribute__((ext_vector_type(2))) float v2f;
typedef __attribute__((ext_vector_type(4))) float v4f;
typedef __attribute__((ext_vector_type(8))) float v8f;
typedef int b128_t __attribute__((vector_size(16)));   // matches builtin proto

#if defined(__gfx1250__) && __has_builtin(__builtin_amdgcn_global_load_async_to_lds_b128)
#define USE_ASYNC 1
#else
#define USE_ASYNC 0
#endif

#if USE_ASYNC
__device__ __forceinline__ void async_cp16(void* lds_dst, const void* gmem_src) {
  // ASYNCcnt-tracked 16B/lane DMA into LDS (global_load_async_to_lds_b128)
  __builtin_amdgcn_global_load_async_to_lds_b128((b128_t*)gmem_src,
                                                 (b128_t*)lds_dst, 0, 0);
}
__device__ __forceinline__ void wait_async0() {
#if __has_builtin(__builtin_amdgcn_s_wait_asynccnt)
  __builtin_amdgcn_s_wait_asynccnt(0);
#else
  asm volatile("s_wait_asynccnt 0" ::: "memory");
#endif
}
#endif

// stage one 64B record: thread t -> raw[t*RECF], gaussian (base + t)
__device__ __forceinline__ void stage_record(float* ldst, const float* gsrc) {
#if USE_ASYNC
#pragma unroll
  for (int q = 0; q < 4; ++q) async_cp16(ldst + q*4, gsrc + q*4);
#else
#pragma unroll
  for (int q = 0; q < 4; ++q) ((v4f*)ldst)[q] = ((const v4f*)gsrc)[q];
#endif
}

// ---------------------------------------------------------------------------
// Kernel 1: stable depth rank (O(G^2), trivial vs. raster cost) + scatter
// sorted records: [a,b,c, mx,my, R, op, cr,cg,cb, 0..0] (64B aligned).
// ---------------------------------------------------------------------------
__global__ __launch_bounds__(256) void rank_sort_kernel(
    const float* __restrict__ means2d, const float* __restrict__ conics,
    const float* __restrict__ colors,  const float* __restrict__ opac,
    const float* __restrict__ depths,  float* __restrict__ recs)
{
  __shared__ float sdep[256];
  const int tid = threadIdx.x;
  const int i = blockIdx.x * 256 + tid;
  const float d = depths[i];
  int rank = 0;
  for (int j0 = 0; j0 < G_N; j0 += 256) {
    sdep[tid] = depths[j0 + tid];
    __syncthreads();
#pragma unroll 8
    for (int jj = 0; jj < 256; ++jj) {
      float dj = sdep[jj];
      int j = j0 + jj;
      rank += (dj < d || (dj == d && j < i)) ? 1 : 0;
    }
    __syncthreads();
  }
  const float a = conics[i*3+0], b = conics[i*3+1], c = conics[i*3+2];
  const float mx = means2d[i*2+0], my = means2d[i*2+1];
  const float op = opac[i];
  // conservative extent: sigma >= 0.5*lmin(conic)*d^2; alpha>=1/255 needs
  // sigma <= ln(255*op)  =>  d <= sqrt(2*ln(255*op)/lmin). +1 px margin.
  const float lmin = 0.5f*((a + c) - sqrtf((a - c)*(a - c) + 4.0f*b*b));
  float R;
  if (op < ALPHA_THRESH) {
    R = -1.0e9f;                       // never visible
  } else {
    float lim = fmaxf(__logf(255.0f*op), 0.0f);
    R = sqrtf(2.0f*lim / fmaxf(lmin, 1e-8f)) + 1.0f;
  }
  float* rec = recs + (size_t)rank * RECF;
  rec[0]=a; rec[1]=b; rec[2]=c; rec[3]=mx; rec[4]=my; rec[5]=R;
  rec[6]=op; rec[7]=colors[i*3+0]; rec[8]=colors[i*3+1]; rec[9]=colors[i*3+2];
#pragma unroll
  for (int k = 10; k < RECF; ++k) rec[k] = 0.0f;
}

// ---------------------------------------------------------------------------
// Kernel 2: tiled rasterizer. Block = 256 threads (8 wave32) = 16x16 tile.
// Wave w owns pixel rows {2w, 2w+1}; lane l -> pixel (x=l&15, row=l>>4).
// ---------------------------------------------------------------------------
__global__ __launch_bounds__(256) void raster_kernel(
    const float* __restrict__ recs, float* __restrict__ out)
{
  __shared__ float  raw[2][CHUNK * RECF];           // double-buffered (32KB)
  __shared__ float  Qc[6][CHUNK];                   // compacted Q vectors
  __shared__ float4 Pk[CHUNK];                      // compacted {op, r, g, b}
  __shared__ unsigned cnts[8];
  __shared__ float  sig[8][2 * TILE * SIG_STRIDE];  // per-wave sigma transpose

  const int tid  = threadIdx.x;
  const int lane = tid & 31;
  const int wv   = tid >> 5;
  const int xm   = lane & 15;      // WMMA M index / pixel x / gaussian column
  const int h    = lane >> 4;      // lane half selects K-pairs & pixel row

  const int bx = blockIdx.x & 31;          // 32x32 tiles
  const int by = blockIdx.x >> 5;
  const float cx = (float)(bx * TILE) + 8.0f;   // tile-local origin (precision)
  const float cy = (float)(by * TILE) + 8.0f;

  // A matrices (P vectors), built once. P = [px^2, px*py, py^2, px, py, 1],
  // px = xm - 7.5 (tile-local center). A layout (16x4 f32, 2 VGPRs):
  // lanes 0-15: v0=K0,v1=K1 ; lanes 16-31: v0=K2,v1=K3.
  const float pxm = (float)xm - 7.5f;
  v2f A1r[2], A2r[2];
#pragma unroll
  for (int r = 0; r < 2; ++r) {
    float pyl = (float)(wv*2 + r) - 7.5f;
    A1r[r].x = h ? pyl*pyl   : pxm*pxm;   // K2 : K0
    A1r[r].y = h ? pxm       : pxm*pyl;   // K3 : K1
    A2r[r].x = h ? 0.0f      : pyl;       // K6 : K4
    A2r[r].y = h ? 0.0f      : 1.0f;      // K7 : K5
  }

  float T = 1.0f, cr = 0.0f, cg = 0.0f, cb = 0.0f;

  float* sb = sig[wv];
  const float* srow = &sig[wv][(h*TILE + xm) * SIG_STRIDE];

  // prologue: stage chunk 0 into buffer 0
  int buf = 0;
  stage_record(&raw[0][tid * RECF], recs + (size_t)tid * RECF);
#if USE_ASYNC
  wait_async0();
#endif

  for (int c0 = 0; c0 < G_N; c0 += CHUNK) {
    // ---- cull + order-preserving compaction (wave32 ballot + scan) ----
    // each thread culls its own staged record (wave-local async wait is enough)
    const float* rc = &raw[buf][tid * RECF];
    const float a = rc[0], b = rc[1], c2 = rc[2];
    const float mxl = rc[3] - cx, myl = rc[4] - cy;
    const float R = rc[5];
    const float opv = rc[6], colr = rc[7], colg = rc[8], colb = rc[9];
    const bool vis = (fabsf(mxl) <= 8.0f + R) && (fabsf(myl) <= 8.0f + R);

    // prefetch next chunk into the other buffer (overlaps with composite)
    if (c0 + CHUNK < G_N)
      stage_record(&raw[buf ^ 1][tid * RECF],
                   recs + (size_t)(c0 + CHUNK + tid) * RECF);

    const unsigned bal = (unsigned)(__ballot(vis) & 0xffffffffull);
    if (lane == 0) cnts[wv] = __popc(bal);
    __syncthreads();
    unsigned off = 0, n = 0;
#pragma unroll
    for (int w2 = 0; w2 < 8; ++w2) {
      unsigned cw = cnts[w2];
      n += cw;
      if (w2 < wv) off += cw;
    }
    if (vis) {
      int p = (int)(off + __popc(bal & ((1u << lane) - 1u)));
      // Q(g) in tile-local coords: sigma = P . Q exactly reproduces
      // 0.5*a*dx^2 + b*dx*dy + 0.5*c*dy^2
      Qc[0][p] = 0.5f * a;
      Qc[1][p] = b;
      Qc[2][p] = 0.5f * c2;
      Qc[3][p] = -(a * mxl + b * myl);
      Qc[4][p] = -(b * mxl + c2 * myl);
      Qc[5][p] = 0.5f * (a * mxl * mxl + c2 * myl * myl) + b * mxl * myl;
      Pk[p] = make_float4(opv, colr, colg, colb);
    }
    if (tid < 16) {                       // pad to group of 16: op=0 => alpha=0
      int npad = ((int)n + 15) & ~15;
      int p = (int)n + tid;
      if (p < npad) {
#pragma unroll
        for (int k = 0; k < 6; ++k) Qc[k][p] = 0.0f;
        Pk[p] = make_float4(0.0f, 0.0f, 0.0f, 0.0f);
      }
    }
    __syncthreads();

    // ---- composite groups of 16 gaussians ----
    for (int g0 = 0; g0 < (int)n; g0 += 16) {
      if (__ballot(T >= TRANS_THRESH) == 0ull) break;   // wave-uniform exit

      const int gi = g0 + xm;                 // B column = gaussian
      v2f B1, B2;                             // B layout: v0=K{0|2}, v1=K{1|3}
      B1.x = Qc[h ? 2 : 0][gi];
      B1.y = Qc[h ? 3 : 1][gi];
      B2.x = h ? 0.0f : Qc[4][gi];
      B2.y = h ? 0.0f : Qc[5][gi];

      // sigma block = P(16px x 6) * Q(6 x 16g), split K = (0..3) + (4..5,pad)
      v8f s0 = {};
      s0 = __builtin_amdgcn_wmma_f32_16x16x4_f32(false, A2r[0], false, B2,
                                                 (short)0, s0, false, false);
      s0 = __builtin_amdgcn_wmma_f32_16x16x4_f32(false, A1r[0], false, B1,
                                                 (short)0, s0, false, false);
      v8f s1 = {};
      s1 = __builtin_amdgcn_wmma_f32_16x16x4_f32(false, A2r[1], false, B2,
                                                 (short)0, s1, false, false);
      s1 = __builtin_amdgcn_wmma_f32_16x16x4_f32(false, A1r[1], false, B1,
                                                 (short)0, s1, false, false);

      // transpose D (lane=gaussian, vgpr=pixel) into per-wave LDS rows
#pragma unroll
      for (int v = 0; v < 8; ++v) {
        int xr = v + 8 * h;                   // D row M = pixel x
        sb[(0*TILE + xr) * SIG_STRIDE + xm] = s0[v];
        sb[(1*TILE + xr) * SIG_STRIDE + xm] = s1[v];
      }
      // same-wave LDS pipe is in-order; wait + block compiler reordering
      asm volatile("s_wait_dscnt 0" ::: "memory");

      // sequential front-to-back compositing for this lane's pixel
#pragma unroll
      for (int k = 0; k < 16; ++k) {
        float4 pk = Pk[g0 + k];               // uniform -> LDS broadcast
        float sg = srow[k];
        float al = fminf(pk.x * __expf(-sg), ALPHA_CAP);
        al = (sg < 0.0f || al < ALPHA_THRESH) ? 0.0f : al;
        float wgt = (T >= TRANS_THRESH) ? al * T : 0.0f;
        cr = fmaf(wgt, pk.y, cr);
        cg = fmaf(wgt, pk.z, cg);
        cb = fmaf(wgt, pk.w, cb);
        T = fmaf(-al, T, T);                  // T *= (1 - alpha)
      }
    }

    // block-wide early exit (also the inter-chunk barrier separating
    // this chunk's Qc/Pk reads from the next chunk's writes)
    if (!__syncthreads_or(T >= TRANS_THRESH ? 1 : 0)) break;

#if USE_ASYNC
    wait_async0();                 // next buffer's own-lane records landed
#endif
    buf ^= 1;
  }

  const int gx = bx * TILE + xm;
  const int gy = by * TILE + wv * 2 + h;
  float* o = out + ((size_t)gy * W_IMG + gx) * 3;
  o[0] = cr; o[1] = cg; o[2] = cb;           // BG == 0, t_rem term vanishes
}

// ---------------------------------------------------------------------------
extern "C" void kernel_launch(void* const* d_in, const int* in_sizes, int n_in,
                              void* d_out, int out_size, void* d_ws, size_t ws_size,
                              hipStream_t stream) {
  (void)in_sizes; (void)n_in; (void)out_size; (void)ws_size;
  const float* means2d = (const float*)d_in[0];
  const float* conics  = (const float*)d_in[1];
  const float* colors  = (const float*)d_in[2];
  const float* opac    = (const float*)d_in[3];
  const float* depths  = (const float*)d_in[4];
  float* recs = (float*)d_ws;                // 8192 * 64B = 512KB scratch

  rank_sort_kernel<<<G_N / 256, 256, 0, stream>>>(means2d, conics, colors,
                                                  opac, depths, recs);
  raster_kernel<<<(W_IMG / TILE) * (H_IMG / TILE), 256, 0, stream>>>(
      recs, (float*)d_out);
}